// PDG2Seq_GCN_1142461301210
// MI455X (gfx1250) — compile-verified
//
#include <hip/hip_runtime.h>

typedef __attribute__((ext_vector_type(2))) float v2f;
typedef __attribute__((ext_vector_type(8))) float v8f;

#define N_NODES 1024
#define CH 64
#define KSTACK 5
#define KC (KSTACK * CH)   // 320 floats per (b,n) row of x_g
#define BATCH 16
#define DEMB 10

// ---------------------------------------------------------------------------
// Copy x[B,N,C] into slot 0 of x_g[B,N,5,C] (stacked layout, row stride 320).
// ---------------------------------------------------------------------------
__global__ __launch_bounds__(256) void copyx_kernel(const float* __restrict__ x,
                                                    float* __restrict__ xg) {
  int f = blockIdx.x * 256 + threadIdx.x;   // float4 id, total B*N*C/4
  int row = f >> 4;                         // flat (b*N + n)
  int c4 = (f & 15) << 2;
  float4 v = reinterpret_cast<const float4*>(x)[f];
  *reinterpret_cast<float4*>(xg + (size_t)row * KC + c4) = v;
}

// ---------------------------------------------------------------------------
// One Chebyshev hop for one support:  Y[b] = A[b] (1024x1024) * X[b] (1024x64)
// X/Y live in x_g slots (row stride KC=320). WG = 4 waves; wave w computes a
// 16x64 strip via V_WMMA_F32_16X16X4_F32.
//
// LDS holds the X panel in k-PAIR-INTERLEAVED layout: lds[p][col][e] with
// k = 2p + e, so a B fragment {K=k, K=k+1} at column col is ONE contiguous
// ds_load_b64 (no 2addr pairing, no packing movs). Double-buffered: panel
// k0+32 is register-staged from global while panel k0 feeds the WMMAs.
// ---------------------------------------------------------------------------
__global__ __launch_bounds__(128) void hop_kernel(const float* __restrict__ adj_s,
                                                  float* xg,
                                                  int src_slot, int dst_slot) {
  __shared__ float lds[2][16 * 128];        // 2 x 8KB : [buf][p][col][e]
  const int b    = blockIdx.y;
  const int tid  = threadIdx.x;
  const int wave = tid >> 5;
  const int lane = tid & 31;
  const int r0   = blockIdx.x * 64 + wave * 16;

  const float* A = adj_s + (size_t)b * N_NODES * N_NODES;
  const float* X = xg + (size_t)b * N_NODES * KC + src_slot * CH;
  float*       Y = xg + (size_t)b * N_NODES * KC + dst_slot * CH;

  // A-fragment: lanes 0-15 -> (M=lane, K=k0,k0+1); lanes 16-31 -> (M=lane-16,
  // K=k0+2,k0+3). One b64 load per lane per k-step.
  const int arow = r0 + (lane & 15);
  const int koff = (lane >> 4) << 1;        // 0 or 2
  const int bcol = lane & 31 & 15;
  const float* Arow = A + (size_t)arow * N_NODES + koff;

  // Staging geometry: thread covers 4 float4 stores; store j (0..511) holds
  // rows k=2p,2p+1 and cols c,c+1 with p=j>>5, c=(j&31)*2.
  v2f ra[4], rb[4];
  auto ldpanel = [&](int k0) {
#pragma unroll
    for (int i = 0; i < 4; ++i) {
      int j = tid + i * 128;
      int p = j >> 5;
      int c = (j & 31) << 1;
      const float* r0p = X + (size_t)(k0 + 2 * p) * KC + c;
      ra[i] = *reinterpret_cast<const v2f*>(r0p);
      rb[i] = *reinterpret_cast<const v2f*>(r0p + KC);
    }
  };
  auto stpanel = [&](int buf) {
#pragma unroll
    for (int i = 0; i < 4; ++i) {
      int j = tid + i * 128;
      int p = j >> 5;
      int c = (j & 31) << 1;
      float4 st = { ra[i].x, rb[i].x, ra[i].y, rb[i].y };   // interleave k-pair
      *reinterpret_cast<float4*>(&lds[buf][p * 128 + (c << 1)]) = st;
    }
  };

  v8f acc0 = {}, acc1 = {}, acc2 = {}, acc3 = {};

  ldpanel(0);
  stpanel(0);
  __syncthreads();

  for (int k0 = 0; k0 < N_NODES; k0 += 32) {
    const int cur = (k0 >> 5) & 1;
    const bool more = (k0 + 32) < N_NODES;
    if (more) {
      ldpanel(k0 + 32);                               // overlap with WMMAs
      __builtin_prefetch(Arow + k0 + 32, 0, 0);       // global_prefetch_b8
    }

    const float* lb = &lds[cur][(koff >> 1) * 128 + (bcol << 1)];
#pragma unroll
    for (int kk = 0; kk < 32; kk += 4) {
      v2f a = *reinterpret_cast<const v2f*>(Arow + k0 + kk);
      const float* lk = lb + (kk << 6);               // (kk/2)*128 floats
      v2f b0 = *reinterpret_cast<const v2f*>(lk + 0);
      v2f b1 = *reinterpret_cast<const v2f*>(lk + 32);
      v2f b2 = *reinterpret_cast<const v2f*>(lk + 64);
      v2f b3 = *reinterpret_cast<const v2f*>(lk + 96);
      acc0 = __builtin_amdgcn_wmma_f32_16x16x4_f32(false, a, false, b0, (short)0, acc0, false, false);
      acc1 = __builtin_amdgcn_wmma_f32_16x16x4_f32(false, a, false, b1, (short)0, acc1, false, false);
      acc2 = __builtin_amdgcn_wmma_f32_16x16x4_f32(false, a, false, b2, (short)0, acc2, false, false);
      acc3 = __builtin_amdgcn_wmma_f32_16x16x4_f32(false, a, false, b3, (short)0, acc3, false, false);
    }

    if (more) stpanel(cur ^ 1);
    __syncthreads();
  }

  // C/D layout: lanes 0-15 -> M=vgpr, lanes 16-31 -> M=vgpr+8; N = lane&15.
  const int mh = (lane >> 4) << 3;
  const int cl = lane & 15;
#pragma unroll
  for (int v = 0; v < 8; ++v) {
    float* yr = Y + (size_t)(r0 + mh + v) * KC + cl;
    yr[0]  = acc0[v];
    yr[16] = acc1[v];
    yr[32] = acc2[v];
    yr[48] = acc3[v];
  }
}

// ---------------------------------------------------------------------------
// Per-node hypernetwork weights, written directly in k-pair-interleaved
// layout  W[n][p][o][e]  (k = 2p + e, p = 0..159, o = 0..63), so the final
// kernel's B fragment is a single global b64 load. Bandwidth-bound (84 MB).
// ---------------------------------------------------------------------------
__global__ __launch_bounds__(256) void weights_kernel(const float* __restrict__ emb,
                                                      const float* __restrict__ wpool,
                                                      float* __restrict__ W) {
  int f = blockIdx.x * 256 + threadIdx.x;   // float4 id, total N*5120
  int n = f / (KC * CH / 4);
  int r = f - n * (KC * CH / 4);
  int p = r >> 5;                           // k-pair 0..159
  int o = (r & 31) << 1;                    // even output col
  const float* wpk = wpool + (p << 7) + o;  // row k=2p, cols o,o+1
  float4 acc = {0.f, 0.f, 0.f, 0.f};
#pragma unroll
  for (int d = 0; d < DEMB; ++d) {
    float e = emb[n * DEMB + d];
    v2f a = *reinterpret_cast<const v2f*>(wpk + (size_t)d * (KC * CH));        // k=2p
    v2f b = *reinterpret_cast<const v2f*>(wpk + (size_t)d * (KC * CH) + CH);   // k=2p+1
    acc.x += e * a.x; acc.y += e * b.x; acc.z += e * a.y; acc.w += e * b.y;
  }
  reinterpret_cast<float4*>(W)[f] = acc;    // {(2p,o),(2p+1,o),(2p,o+1),(2p+1,o+1)}
}

// ---------------------------------------------------------------------------
// Per-node contraction: out[0:16,n,0:64] = x_g[0:16,n,:320] * W[n] + bias[n].
// 16-row WMMA dimension = batch axis (B == 16). One WG (4 waves) per node;
// wave ct owns output columns [16*ct, 16*ct+16). B frags are single b64 loads
// thanks to the interleaved W layout.
// ---------------------------------------------------------------------------
__global__ __launch_bounds__(128) void final_kernel(const float* __restrict__ xg,
                                                    const float* __restrict__ W,
                                                    const float* __restrict__ emb,
                                                    const float* __restrict__ bpool,
                                                    float* __restrict__ out) {
  const int n    = blockIdx.x;
  const int tid  = threadIdx.x;
  const int ct   = tid >> 5;
  const int lane = tid & 31;
  const int koff = (lane >> 4) << 1;
  const int cl   = lane & 15;

  const float* Arow = xg + ((size_t)cl * N_NODES + n) * KC + koff;  // row = batch b
  const float* Bp   = W + (size_t)n * (KC * CH)
                        + (koff >> 1) * 128 + ((ct * 16 + cl) << 1);

  v8f acc = {};
#pragma unroll 8
  for (int kk = 0; kk < KC; kk += 4) {
    v2f a  = *reinterpret_cast<const v2f*>(Arow + kk);
    v2f bf = *reinterpret_cast<const v2f*>(Bp + (kk << 6));   // (kk/2)*128 floats
    acc = __builtin_amdgcn_wmma_f32_16x16x4_f32(false, a, false, bf, (short)0, acc, false, false);
  }

  const int o = ct * 16 + cl;
  float bv = 0.f;
#pragma unroll
  for (int d = 0; d < DEMB; ++d)
    bv += emb[n * DEMB + d] * bpool[d * CH + o];

  const int mh = (lane >> 4) << 3;
#pragma unroll
  for (int v = 0; v < 8; ++v)
    out[((size_t)(mh + v) * N_NODES + n) * CH + o] = acc[v] + bv;
}

// ---------------------------------------------------------------------------
extern "C" void kernel_launch(void* const* d_in, const int* in_sizes, int n_in,
                              void* d_out, int out_size, void* d_ws, size_t ws_size,
                              hipStream_t stream) {
  const float* x     = (const float*)d_in[0];   // [16,1024,64]
  const float* adj   = (const float*)d_in[1];   // [2,16,1024,1024]
  const float* emb   = (const float*)d_in[2];   // [1024,10]
  const float* wpool = (const float*)d_in[3];   // [10,5,64,64]
  const float* bpool = (const float*)d_in[4];   // [10,64]
  float* out = (float*)d_out;                   // [16,1024,64]

  float* xg = (float*)d_ws;                         // x_g [16,1024,5,64] : 21 MB
  float* W  = xg + (size_t)BATCH * N_NODES * KC;    // W  [1024,160,64,2] : 84 MB

  copyx_kernel<<<(BATCH * N_NODES * CH / 4) / 256, 256, 0, stream>>>(x, xg);

  dim3 hgrid(N_NODES / 64, BATCH);
  const size_t supp = (size_t)BATCH * N_NODES * N_NODES;
  hop_kernel<<<hgrid, 128, 0, stream>>>(adj,        xg, 0, 1);  // support 0, hop 1
  hop_kernel<<<hgrid, 128, 0, stream>>>(adj,        xg, 1, 2);  // support 0, hop 2
  hop_kernel<<<hgrid, 128, 0, stream>>>(adj + supp, xg, 0, 3);  // support 1, hop 1
  hop_kernel<<<hgrid, 128, 0, stream>>>(adj + supp, xg, 3, 4);  // support 1, hop 2

  weights_kernel<<<(N_NODES * (KC * CH / 4)) / 256, 256, 0, stream>>>(emb, wpool, W);
  final_kernel<<<N_NODES, 128, 0, stream>>>(xg, W, emb, bpool, out);
}